// GCN_34153579938095
// MI455X (gfx1250) — compile-verified
//
#include <hip/hip_runtime.h>

typedef __attribute__((ext_vector_type(2))) float v2f;
typedef __attribute__((ext_vector_type(8))) float v8f;

#define N_NODES  100000
#define N_EDGES  3200000
#define IN_CH    128
#define HID      64
#define N_GRAPHS 512

// ---------------------------------------------------------------- utilities
__global__ void k_zero(float* __restrict__ p, int n) {
  int i = blockIdx.x * 256 + threadIdx.x;
  if (i < n) p[i] = 0.0f;
}

// deg[dst] += 1 over all edges (f32 atomics; deg buffer pre-zeroed)
__global__ void k_deg(const int* __restrict__ dst, float* __restrict__ deg, int E) {
  int e = blockIdx.x * 256 + threadIdx.x;
  if (e < E) atomicAdd(&deg[dst[e]], 1.0f);
}

// in place: dis[i] = rsqrt(deg[i] + 1)   (self-loop included)
__global__ void k_rsqrt(float* __restrict__ d, int n) {
  int i = blockIdx.x * 256 + threadIdx.x;
  if (i < n) d[i] = rsqrtf(d[i] + 1.0f);
}

// ------------------------------------------------- WMMA f32 GEMM: Y = X @ W
// X:[M,K] row-major, W:[K,64] row-major, Y:[M,64]. M%16==0, K%4==0.
// One wave32 computes one 16x16 tile via V_WMMA_F32_16X16X4_F32.
__global__ void k_gemm_wmma(const float* __restrict__ X, const float* __restrict__ W,
                            float* __restrict__ Y, int K) {
  int tile = blockIdx.x * 8 + (threadIdx.x >> 5);   // 8 waves per block
  int lane = threadIdx.x & 31;
  int tm = tile >> 2;          // row tile (16 rows)
  int tn = tile & 3;           // col tile (4 tiles of 16 over N=64)
  int half = (lane < 16) ? 0 : 1;
  int l15  = lane & 15;

  const float* Xr = X + (long)(tm * 16 + l15) * K;  // A: row per lane
  const float* Wc = W + tn * 16 + l15;              // B: column per lane
  int ka = half * 2;                                // lanes 16-31 hold K+2,K+3

  v8f c = {};
  for (int k = 0; k < K; k += 4) {
    v2f a = *(const v2f*)(Xr + k + ka);             // A[m][k+ka], A[m][k+ka+1]
    v2f b;
    b.x = Wc[(k + ka)     * HID];                   // B[k+ka][n]
    b.y = Wc[(k + ka + 1) * HID];                   // B[k+ka+1][n]
    c = __builtin_amdgcn_wmma_f32_16x16x4_f32(
        /*neg_a=*/false, a, /*neg_b=*/false, b,
        /*c_mod=*/(short)0, c, /*reuse_a=*/false, /*reuse_b=*/false);
  }

  int mbase = tm * 16 + half * 8;                   // VGPR v -> M = v or v+8
  int n     = tn * 16 + l15;
#pragma unroll
  for (int v = 0; v < 8; ++v)
    Y[(long)(mbase + v) * HID + n] = c[v];
}

// ---------------------------------------------- GCN aggregation components
// out[i][c] = h[i][c]*dis[i]^2 + b[c]     (self-loop term + bias, pre-scatter)
__global__ void k_self_init(const float* __restrict__ h, const float* __restrict__ dis,
                            const float* __restrict__ b, float* __restrict__ out, int n) {
  int i = blockIdx.x * 256 + threadIdx.x;
  if (i >= n) return;
  float di = dis[i >> 6];
  out[i] = h[i] * di * di + b[i & 63];
}

// per edge: out[dst] += h[src] * (dis[src]*dis[dst]); one wave32 per edge,
// each lane moves 2 channels (float2 gather of the contiguous 256B row).
__global__ void k_edge_scatter(const int* __restrict__ src, const int* __restrict__ dst,
                               const float* __restrict__ dis, const float* __restrict__ h,
                               float* __restrict__ out, int E) {
  int e = blockIdx.x * 8 + (threadIdx.x >> 5);
  if (e >= E) return;
  int lane = threadIdx.x & 31;
  int s = src[e];
  int d = dst[e];
  float nrm = dis[s] * dis[d];
  v2f v = ((const v2f*)(h + (long)s * HID))[lane];
  float* o = out + (long)d * HID + lane * 2;
  atomicAdd(o,     v.x * nrm);
  atomicAdd(o + 1, v.y * nrm);
}

__global__ void k_relu(float* __restrict__ p, int n) {
  int i = blockIdx.x * 256 + threadIdx.x;
  if (i < n) p[i] = fmaxf(p[i], 0.0f);
}

// ---------------------------------------------------------------- pooling
__global__ void k_cnt(const int* __restrict__ batch, float* __restrict__ cnt, int n) {
  int i = blockIdx.x * 256 + threadIdx.x;
  if (i < n) atomicAdd(&cnt[batch[i]], 1.0f);
}

__global__ void k_pool(const float* __restrict__ h, const int* __restrict__ batch,
                       float* __restrict__ out, int n) {
  int i = blockIdx.x * 256 + threadIdx.x;
  if (i >= n) return;
  int node = i >> 6;
  atomicAdd(&out[(long)batch[node] * HID + (i & 63)], h[i]);
}

__global__ void k_div(float* __restrict__ out, const float* __restrict__ cnt, int n) {
  int i = blockIdx.x * 256 + threadIdx.x;
  if (i < n) out[i] = out[i] / fmaxf(cnt[i >> 6], 1.0f);
}

// ---------------------------------------------------------------- driver
extern "C" void kernel_launch(void* const* d_in, const int* in_sizes, int n_in,
                              void* d_out, int out_size, void* d_ws, size_t ws_size,
                              hipStream_t stream) {
  const float* x     = (const float*)d_in[0];
  const int*   ei    = (const int*)d_in[1];   // [2, E] flat: [src..., dst...]
  const int*   batch = (const int*)d_in[2];
  const float* W1    = (const float*)d_in[3];
  const float* b1    = (const float*)d_in[4];
  const float* W2    = (const float*)d_in[5];
  const float* b2    = (const float*)d_in[6];
  float* out = (float*)d_out;
  float* ws  = (float*)d_ws;

  const int* srcE = ei;
  const int* dstE = ei + N_EDGES;

  // workspace layout (floats)
  float* dis = ws;                          // N_NODES (deg -> dis in place)
  float* hA  = ws + 100352;                 // N_NODES*HID (aligned start)
  float* hB  = hA + (size_t)N_NODES * HID;  // N_NODES*HID
  float* cnt = hB + (size_t)N_NODES * HID;  // N_GRAPHS

  dim3 blk(256);
  const int NH = N_NODES * HID;
  const int gemm_blocks = (N_NODES / 16) * 4 / 8;   // 3125 (exact)
  const int edge_blocks = N_EDGES / 8;              // 400000 (exact)

  // normalized degrees: dis[i] = (deg_i + 1)^{-1/2}
  k_zero <<<(N_NODES + 255) / 256, blk, 0, stream>>>(dis, N_NODES);
  k_deg  <<<(N_EDGES + 255) / 256, blk, 0, stream>>>(dstE, dis, N_EDGES);
  k_rsqrt<<<(N_NODES + 255) / 256, blk, 0, stream>>>(dis, N_NODES);

  // ---- layer 1: h = relu( scatter(norm * (x@W1)[src]) + (x@W1)*dis^2 + b1 )
  k_gemm_wmma   <<<gemm_blocks, blk, 0, stream>>>(x, W1, hA, IN_CH);
  k_self_init   <<<(NH + 255) / 256, blk, 0, stream>>>(hA, dis, b1, hB, NH);
  k_edge_scatter<<<edge_blocks, blk, 0, stream>>>(srcE, dstE, dis, hA, hB, N_EDGES);
  k_relu        <<<(NH + 255) / 256, blk, 0, stream>>>(hB, NH);

  // ---- layer 2
  k_gemm_wmma   <<<gemm_blocks, blk, 0, stream>>>(hB, W2, hA, HID);
  k_self_init   <<<(NH + 255) / 256, blk, 0, stream>>>(hA, dis, b2, hB, NH);
  k_edge_scatter<<<edge_blocks, blk, 0, stream>>>(srcE, dstE, dis, hA, hB, N_EDGES);
  k_relu        <<<(NH + 255) / 256, blk, 0, stream>>>(hB, NH);

  // ---- global mean pool over batch ids
  k_zero<<<(N_GRAPHS * HID + 255) / 256, blk, 0, stream>>>(out, N_GRAPHS * HID);
  k_zero<<<(N_GRAPHS + 255) / 256, blk, 0, stream>>>(cnt, N_GRAPHS);
  k_cnt <<<(N_NODES + 255) / 256, blk, 0, stream>>>(batch, cnt, N_NODES);
  k_pool<<<(NH + 255) / 256, blk, 0, stream>>>(hB, batch, out, NH);
  k_div <<<(N_GRAPHS * HID + 255) / 256, blk, 0, stream>>>(out, cnt, N_GRAPHS * HID);
}